// MimoLinearDynamicalOperator_12841952215803
// MI455X (gfx1250) — compile-verified
//
#include <hip/hip_runtime.h>

// MIMO linear dynamical operator: FIR (im2col GEMM via V_WMMA_F32_16X16X4_F32)
// followed by an in-place per-channel order-8 AR scan.
//
// Shapes (from reference): B=16, T=4096, IN_CH=64, OUT_CH=64, N_B=16, N_A=8.

typedef __attribute__((ext_vector_type(2))) float v2f;
typedef __attribute__((ext_vector_type(8))) float v8f;

#define T_LEN   4096
#define IN_CH   64
#define OUT_CH  64
#define NB      16
#define NA      8
#define TILE_T  32                 // t-rows of output per block
#define STAGE_ROWS (TILE_T + NB - 1)   // 47 staged x rows (incl. 15 history)
#define LDS_STRIDE 49              // odd stride -> conflict-free staging

// ---------------------------------------------------------------------------
// FIR stage: u[b,t,n] = sum_{w=0..15} sum_{i=0..63} x[b,t+w-15,i] * W[w,i,n]
// One wave computes a 16x16 (t x n) tile; 8 waves/block -> 32x64 tile.
// K = NB*IN_CH = 1024, consumed 4 at a time by v_wmma_f32_16x16x4_f32.
// ---------------------------------------------------------------------------
__global__ __launch_bounds__(256) void fir_wmma_kernel(
    const float* __restrict__ x,   // [B, T, IN_CH]
    const float* __restrict__ w,   // [NB, IN_CH, OUT_CH] viewed as [1024, 64]
    float* __restrict__ u)         // [B, T, OUT_CH]  (written fully)
{
    __shared__ float xs[IN_CH][LDS_STRIDE];   // xs[i][local_t], 12.5 KB

    const int tid  = threadIdx.x;
    const int bpb  = T_LEN / TILE_T;          // blocks per batch = 128
    const int b    = blockIdx.x / bpb;
    const int t0   = (blockIdx.x % bpb) * TILE_T;

    // Warm L2 with the (tiny, fully shared) weight matrix: 1024 rows x 256 B.
    for (int row = tid; row < NB * IN_CH; row += 256)
        __builtin_prefetch(w + (size_t)row * OUT_CH, 0, 1);

    // Stage x[b, t0-15 .. t0+31, :] into LDS transposed (i-major), zeros for t<0.
    {
        const int ci = tid & (IN_CH - 1);
        for (int r = tid >> 6; r < STAGE_ROWS; r += 4) {
            const int tg = t0 - (NB - 1) + r;
            float v = 0.0f;
            if (tg >= 0)
                v = x[((size_t)b * T_LEN + tg) * IN_CH + ci];
            xs[ci][r] = v;
        }
    }
    __syncthreads();

    const int lane  = tid & 31;
    const int wave  = tid >> 5;
    const int mtile = wave >> 2;              // 0..1 : which 16 t-rows
    const int ntile = wave & 3;               // 0..3 : which 16 out channels
    const int mrow  = lane & 15;              // M index (A rows / C cols map)
    const int hi    = lane >> 4;              // lane half: K offset +2
    const int ncol  = ntile * 16 + (lane & 15);

    v8f c = {};
    for (int tap = 0; tap < NB; ++tap) {
        // staged row for output t = t0 + mtile*16 + mrow, shifted by tap:
        // local_t = (t + tap - 15) - (t0 - 15) = mtile*16 + mrow + tap
        const int tl = mtile * 16 + mrow + tap;
        const float* wk = w + ((size_t)tap * IN_CH) * OUT_CH + ncol;
#pragma unroll
        for (int i4 = 0; i4 < IN_CH; i4 += 4) {
            const int kk = i4 + hi * 2;       // ISA 32-bit A layout: lanes>=16 hold K+2
            v2f a, bf;
            a.x  = xs[kk][tl];                // A(m, kk)   -> VGPR0
            a.y  = xs[kk + 1][tl];            // A(m, kk+1) -> VGPR1
            bf.x = wk[(size_t)kk * OUT_CH];       // B(kk,   n)
            bf.y = wk[(size_t)(kk + 1) * OUT_CH]; // B(kk+1, n)
            c = __builtin_amdgcn_wmma_f32_16x16x4_f32(
                    /*neg_a=*/false, a, /*neg_b=*/false, bf,
                    /*c_mod=*/(short)0, c, /*reuse_a=*/false, /*reuse_b=*/false);
        }
    }

    // C layout: VGPR v holds row M = v + 8*hi, N = lane&15.
    const int trow = t0 + mtile * 16 + hi * 8;
    float* uout = u + ((size_t)b * T_LEN) * OUT_CH + ncol;
#pragma unroll
    for (int v = 0; v < 8; ++v)
        uout[(size_t)(trow + v) * OUT_CH] = c[v];
}

// ---------------------------------------------------------------------------
// AR stage (in place on d_out): y[t] = u[t] + sum_{j=1..8} a[j-1,o] * y[t-j]
// One thread per (b, o) chain; reads are coalesced across o.
// ---------------------------------------------------------------------------
__global__ __launch_bounds__(64) void ar_scan_kernel(
    const float* __restrict__ a_coeff,   // [NA, OUT_CH]
    float* __restrict__ y)               // [B, T, OUT_CH], holds u on entry
{
    const int b = blockIdx.x;
    const int o = threadIdx.x;

    float a[NA];
#pragma unroll
    for (int j = 0; j < NA; ++j)
        a[j] = a_coeff[j * OUT_CH + o];

    float h[NA];                          // h[j] = y[t-1-j]
#pragma unroll
    for (int j = 0; j < NA; ++j) h[j] = 0.0f;

    float* p = y + ((size_t)b * T_LEN) * OUT_CH + o;

#pragma unroll 8
    for (int t = 0; t < T_LEN; ++t) {
        float acc = p[(size_t)t * OUT_CH];
        // accumulate oldest->newest so the final FMA (on y[t-1]) is the chain dep
#pragma unroll
        for (int j = NA - 1; j >= 0; --j)
            acc = fmaf(a[j], h[j], acc);
#pragma unroll
        for (int j = NA - 1; j > 0; --j) h[j] = h[j - 1];
        h[0] = acc;
        p[(size_t)t * OUT_CH] = acc;
    }
}

// ---------------------------------------------------------------------------
extern "C" void kernel_launch(void* const* d_in, const int* in_sizes, int n_in,
                              void* d_out, int out_size, void* d_ws, size_t ws_size,
                              hipStream_t stream)
{
    const float* x = (const float*)d_in[0];   // inputs  [B, T, 64]
    const float* a = (const float*)d_in[1];   // a_coeff [8, 64]
    const float* w = (const float*)d_in[2];   // b_coeff [16, 64, 64]
    float* out = (float*)d_out;               // [B, T, 64]

    const int B = in_sizes[0] / (T_LEN * IN_CH);

    // FIR GEMM via fp32 WMMA, writes u directly into d_out.
    dim3 grid1((unsigned)(B * (T_LEN / TILE_T)));
    fir_wmma_kernel<<<grid1, 256, 0, stream>>>(x, w, out);

    // AR scan in place on d_out (reads u[t] before overwriting with y[t]).
    ar_scan_kernel<<<dim3((unsigned)B), 64, 0, stream>>>(a, out);
}